// EventDecoder_36249523978295
// MI455X (gfx1250) — compile-verified
//
#include <hip/hip_runtime.h>
#include <hip/hip_bf16.h>

#define NNODES 262144
#define NBATCH 64
#define FDIM   320      // C*D = 5*64
#define NCHUNK 16       // row chunks per segment
#define EOUT   3
#define KDIM   960      // 3*FDIM

typedef __attribute__((ext_vector_type(2))) float v2f;
typedef __attribute__((ext_vector_type(4))) float v4f;
typedef __attribute__((ext_vector_type(8))) float v8f;

#define NEG_HUGE (-__FLT_MAX__)

// ---------------------------------------------------------------------------
// Kernel 1: segment boundaries via binary search (batch ids are sorted).
// bounds[p*65 + s] = first row index with batch >= s;  bounds[p*65+64] = N.
// ---------------------------------------------------------------------------
__global__ void bounds_kernel(const int* __restrict__ bu,
                              const int* __restrict__ bv,
                              const int* __restrict__ by,
                              int* __restrict__ bounds) {
    int tid = blockIdx.x * blockDim.x + threadIdx.x;
    if (tid >= 3 * (NBATCH + 1)) return;
    int p = tid / (NBATCH + 1);
    int s = tid % (NBATCH + 1);
    const int* arr = (p == 0) ? bu : (p == 1) ? bv : by;
    int lo = 0, hi = NNODES;
    while (lo < hi) {
        int mid = (lo + hi) >> 1;
        if (arr[mid] < s) lo = mid + 1; else hi = mid;
    }
    bounds[p * (NBATCH + 1) + s] = lo;
}

// ---------------------------------------------------------------------------
// Kernel 2: one-pass online softmax aggregation (flash style) over a row
// chunk of one (plane, segment). 320 threads = 4 row substreams x 80 float4
// lanes; each thread keeps running (max, denom, sum e*x) per feature.
// Partials per (plane, seg, chunk, feature) go to workspace.
// ---------------------------------------------------------------------------
__global__ __launch_bounds__(320)
void aggr_kernel(const float* __restrict__ mu, const float* __restrict__ mv,
                 const float* __restrict__ my,
                 const float* __restrict__ tu, const float* __restrict__ tv,
                 const float* __restrict__ ty,
                 const int*   __restrict__ bounds,
                 float* __restrict__ pM, float* __restrict__ pD,
                 float* __restrict__ pA) {
    const int p = blockIdx.z;
    const int s = blockIdx.y;
    const int c = blockIdx.x;

    const float* __restrict__ mp = (p == 0) ? mu : (p == 1) ? mv : my;
    const float  t = ((p == 0) ? tu : (p == 1) ? tv : ty)[0];

    const int r0  = bounds[p * (NBATCH + 1) + s];
    const int r1  = bounds[p * (NBATCH + 1) + s + 1];
    const int len = r1 - r0;
    const int per = (len + NCHUNK - 1) / NCHUNK;
    const int cs  = r0 + c * per;
    const int ce  = min(cs + per, r1);

    const int j = threadIdx.x / 80;  // row substream 0..3
    const int q = threadIdx.x % 80;  // float4 index 0..79 (features 4q..4q+3)

    v4f mx = {NEG_HUGE, NEG_HUGE, NEG_HUGE, NEG_HUGE};
    v4f dn = {0.f, 0.f, 0.f, 0.f};
    v4f ac = {0.f, 0.f, 0.f, 0.f};

    #pragma unroll 2
    for (int r = cs + j; r < ce; r += 4) {
        const v4f* src = (const v4f*)(mp + (size_t)r * FDIM + q * 4);
        v4f x = __builtin_nontemporal_load(src);   // 1GB stream: NT, skip cache
        #pragma unroll
        for (int k = 0; k < 4; ++k) {
            float sv = t * x[k];
            float nm = fmaxf(mx[k], sv);
            float sc = __expf(mx[k] - nm);   // 0 on first element
            float e  = __expf(sv - nm);
            dn[k] = dn[k] * sc + e;
            ac[k] = ac[k] * sc + e * x[k];
            mx[k] = nm;
        }
    }

    // merge the 4 row substreams through LDS
    __shared__ float sm[4 * FDIM];
    __shared__ float sd[4 * FDIM];
    __shared__ float sa[4 * FDIM];
    #pragma unroll
    for (int k = 0; k < 4; ++k) {
        sm[j * FDIM + 4 * q + k] = mx[k];
        sd[j * FDIM + 4 * q + k] = dn[k];
        sa[j * FDIM + 4 * q + k] = ac[k];
    }
    __syncthreads();

    const int f = threadIdx.x;            // 320 threads == 320 features
    float M = NEG_HUGE, D = 0.f, A = 0.f;
    #pragma unroll
    for (int jj = 0; jj < 4; ++jj) {
        float mj = sm[jj * FDIM + f];
        float dj = sd[jj * FDIM + f];
        float aj = sa[jj * FDIM + f];
        if (dj != 0.f) {
            float nm = fmaxf(M, mj);
            float s0 = __expf(M - nm);
            float s1 = __expf(mj - nm);
            D = D * s0 + dj * s1;
            A = A * s0 + aj * s1;
            M = nm;
        }
    }
    size_t o = (((size_t)(p * NBATCH + s)) * NCHUNK + c) * FDIM + f;
    pM[o] = M; pD[o] = D; pA[o] = A;
}

// ---------------------------------------------------------------------------
// Kernel 3: merge the NCHUNK chunk partials -> feat[64, 960]
// (plane p occupies feature columns p*320 .. p*320+319)
// ---------------------------------------------------------------------------
__global__ __launch_bounds__(FDIM)
void combine_kernel(const float* __restrict__ pM, const float* __restrict__ pD,
                    const float* __restrict__ pA, float* __restrict__ feat) {
    const int s = blockIdx.x;
    const int p = blockIdx.y;
    const int f = threadIdx.x;
    float M = NEG_HUGE, D = 0.f, A = 0.f;
    for (int c = 0; c < NCHUNK; ++c) {
        size_t o = (((size_t)(p * NBATCH + s)) * NCHUNK + c) * FDIM + f;
        float mj = pM[o], dj = pD[o], aj = pA[o];
        if (dj != 0.f) {
            float nm = fmaxf(M, mj);
            float s0 = __expf(M - nm);
            float s1 = __expf(mj - nm);
            D = D * s0 + dj * s1;
            A = A * s0 + aj * s1;
            M = nm;
        }
    }
    feat[(size_t)s * KDIM + p * FDIM + f] = (D != 0.f) ? (A / D) : 0.f;
}

// ---------------------------------------------------------------------------
// Kernel 4: out[64,3] = feat[64,960] @ W[3,960]^T + b via V_WMMA_F32_16X16X4_F32.
// One block, 4 waves; wave w owns output rows 16w..16w+15 (N padded to 16).
// A layout (16x4 f32): lane m = l%16, VGPR v holds K = (l/16)*2 + v.
// B layout (4x16 f32): lane n = l%16, VGPR v holds K = (l/16)*2 + v.
// C/D (16x16 f32): VGPR r -> M = r + 8*(l/16), N = l%16.
// ---------------------------------------------------------------------------
__global__ __launch_bounds__(128)
void gemm_wmma_kernel(const float* __restrict__ feat, const float* __restrict__ W,
                      const float* __restrict__ bias, float* __restrict__ out) {
    __shared__ float Wl[KDIM * 16];           // W^T zero-padded to 16 cols
    for (int i = threadIdx.x; i < KDIM * 16; i += 128) {
        int k = i >> 4, n = i & 15;
        Wl[i] = (n < EOUT) ? W[n * KDIM + k] : 0.f;
    }
    __syncthreads();

    const int w    = threadIdx.x >> 5;        // wave 0..3
    const int l    = threadIdx.x & 31;        // lane in wave
    const int half = l >> 4;                  // 0: lanes 0-15, 1: lanes 16-31
    const int kb   = half * 2;
    const int n    = l & 15;
    const int mrow = w * 16 + (l & 15);

    v8f acc = {};
    for (int k = 0; k < KDIM; k += 4) {
        v2f a, b;
        a.x = feat[(size_t)mrow * KDIM + k + kb];
        a.y = feat[(size_t)mrow * KDIM + k + kb + 1];
        b.x = Wl[(k + kb) * 16 + n];
        b.y = Wl[(k + kb + 1) * 16 + n];
        acc = __builtin_amdgcn_wmma_f32_16x16x4_f32(
            /*neg_a=*/false, a, /*neg_b=*/false, b,
            /*c_mod=*/(short)0, acc, /*reuse_a=*/false, /*reuse_b=*/false);
    }

    #pragma unroll
    for (int r = 0; r < 8; ++r) {
        int m = w * 16 + r + 8 * half;
        if (n < EOUT) out[m * EOUT + n] = acc[r] + bias[n];
    }
}

// ---------------------------------------------------------------------------
extern "C" void kernel_launch(void* const* d_in, const int* in_sizes, int n_in,
                              void* d_out, int out_size, void* d_ws, size_t ws_size,
                              hipStream_t stream) {
    const float* mu = (const float*)d_in[0];
    const int*   bu = (const int*)  d_in[1];
    const float* tu = (const float*)d_in[2];
    const float* mv = (const float*)d_in[3];
    const int*   bv = (const int*)  d_in[4];
    const float* tv = (const float*)d_in[5];
    const float* my = (const float*)d_in[6];
    const int*   by = (const int*)  d_in[7];
    const float* ty = (const float*)d_in[8];
    const float* W  = (const float*)d_in[9];
    const float* b  = (const float*)d_in[10];

    float* ws = (float*)d_ws;
    const size_t P = (size_t)3 * NBATCH * NCHUNK * FDIM;   // partials per array
    float* pM    = ws;
    float* pD    = ws + P;
    float* pA    = ws + 2 * P;
    float* feat  = ws + 3 * P;                             // [64, 960]
    int*   bnd   = (int*)(feat + (size_t)NBATCH * KDIM);   // [3, 65]

    bounds_kernel<<<4, 64, 0, stream>>>(bu, bv, by, bnd);

    aggr_kernel<<<dim3(NCHUNK, NBATCH, 3), 320, 0, stream>>>(
        mu, mv, my, tu, tv, ty, bnd, pM, pD, pA);

    combine_kernel<<<dim3(NBATCH, 3), FDIM, 0, stream>>>(pM, pD, pA, feat);

    gemm_wmma_kernel<<<1, 128, 0, stream>>>(feat, W, b, (float*)d_out);
}